// SingleVolumeRenderer_80925773791499
// MI455X (gfx1250) — compile-verified
//
#include <hip/hip_runtime.h>
#include <hip/hip_bf16.h>
#include <math.h>

#define EPS_DIV  1e-10f
#define EPS_NORM 1e-12f

// ---------------------------------------------------------------------------
// gfx1250 async-copy-to-LDS path (ASYNCcnt), guarded so we always compile.
// Probe feedback: __builtin_amdgcn_global_load_async_to_lds_b32 takes
// (AS1 int* gaddr, AS3 int* ldsaddr, imm offset, imm cpol).
// ---------------------------------------------------------------------------
#if defined(__has_builtin)
#  if __has_builtin(__builtin_amdgcn_global_load_async_to_lds_b32) && \
      __has_builtin(__builtin_amdgcn_s_wait_asynccnt)
#    define USE_ASYNC_LDS 1
#  endif
#endif
#ifndef USE_ASYNC_LDS
#  define USE_ASYNC_LDS 0
#endif

#if USE_ASYNC_LDS
typedef __attribute__((address_space(1))) int ga_i32;
typedef __attribute__((address_space(3))) int ls_i32;
#endif

// ---------------------------------------------------------------------------
// Zero-fill the [N_rays, 8] output (non-hit rays must be 0; d_out is poisoned).
// ---------------------------------------------------------------------------
__global__ __launch_bounds__(256) void vr_zero_kernel(float4* __restrict__ out4, int n4) {
    int i = blockIdx.x * blockDim.x + threadIdx.x;
    if (i < n4) out4[i] = make_float4(0.f, 0.f, 0.f, 0.f);
}

// ---------------------------------------------------------------------------
// One wave32 per hit ray. Per-segment exclusive cumsum of log1p(-alpha) via
// wave shuffle scan (carry across 32-sample chunks), then weighted segment
// reductions; lane 0 writes 8 floats (two float4s) to out[ray*8].
// alpha (the scan-critical operand) is software-pipelined one chunk ahead
// through LDS with the gfx1250 async-copy engine when available.
// ---------------------------------------------------------------------------
__global__ __launch_bounds__(256) void vr_render_kernel(
    const float* __restrict__ alpha,
    const float* __restrict__ tdep,
    const float* __restrict__ rgb,
    const float* __restrict__ nablas,
    const int*   __restrict__ ray_inds_hit,
    const int*   __restrict__ starts,
    float*       __restrict__ out,
    int n_hit, int P)
{
#if USE_ASYNC_LDS
    __shared__ float s_alpha[8][2][32];           // per-wave double buffer (2 KB)
#endif
    const int lane = threadIdx.x & 31;
    const int wib  = threadIdx.x >> 5;            // wave index in block (0..7)
    const int r    = blockIdx.x * 8 + wib;        // hit-ray index (one wave per ray)
    if (r >= n_hit) return;

    const int s = starts[r];
    const int e = (r + 1 < n_hit) ? starts[r + 1] : P;
    const int nchunks = (e - s + 31) >> 5;        // may be 0 (empty segment)

    float carry  = 0.f;                            // running cumsum of log1p(-a)
    float acc_vw = 0.f, acc_vwt = 0.f;
    float acc_r0 = 0.f, acc_r1 = 0.f, acc_r2 = 0.f;
    float acc_n0 = 0.f, acc_n1 = 0.f, acc_n2 = 0.f;

#if USE_ASYNC_LDS
    if (nchunks > 0) {                             // prologue: stage chunk 0
        const int i0 = s + lane;
        if (i0 < e)
            __builtin_amdgcn_global_load_async_to_lds_b32(
                (ga_i32*)(alpha + i0), (ls_i32*)&s_alpha[wib][0][lane], 0, 0);
    }
#endif

    for (int c = 0; c < nchunks; ++c) {
        const int  base = s + (c << 5);
        const int  i    = base + lane;
        const bool act  = (i < e);

        // Prefetch next chunk of the streaming operands (global_prefetch_b8;
        // speculative, silently dropped if the address is invalid).
        __builtin_prefetch(tdep   + i + 32,       0, 0);
        __builtin_prefetch(rgb    + 3 * (i + 32), 0, 0);
        __builtin_prefetch(nablas + 3 * (i + 32), 0, 0);

#if USE_ASYNC_LDS
        if (c + 1 < nchunks) {                     // stage chunk c+1 async
            const int j = i + 32;
            if (j < e)
                __builtin_amdgcn_global_load_async_to_lds_b32(
                    (ga_i32*)(alpha + j),
                    (ls_i32*)&s_alpha[wib][(c + 1) & 1][lane], 0, 0);
            __builtin_amdgcn_s_wait_asynccnt(1);   // chunk c landed, c+1 in flight
        } else {
            __builtin_amdgcn_s_wait_asynccnt(0);   // last chunk: drain
        }
        const float a = act ? s_alpha[wib][c & 1][lane] : 0.f;
#else
        __builtin_prefetch(alpha + i + 32, 0, 0);
        const float a = act ? __builtin_nontemporal_load(alpha + i) : 0.f;
#endif
        const float ac = fminf(fmaxf(a, 0.f), 1.f - 1e-7f);
        const float l  = act ? log1pf(-ac) : 0.f;  // inactive lanes: scan identity

        // Inclusive wave32 scan of l.
        float incl = l;
        #pragma unroll
        for (int off = 1; off < 32; off <<= 1) {
            float v = __shfl_up(incl, off, 32);
            if (lane >= off) incl += v;
        }
        const float total = __shfl(incl, 31, 32);  // chunk total -> carry
        const float excl  = incl - l;              // exclusive scan
        const float trans = expf(carry + excl);    // transmittance
        carry += total;

        if (act) {
            const float vw = a * trans;            // reference multiplies UNclipped alpha
            acc_vw  += vw;
            acc_vwt += vw * __builtin_nontemporal_load(tdep + i);

            const float c0 = __builtin_nontemporal_load(rgb + 3 * i + 0);
            const float c1 = __builtin_nontemporal_load(rgb + 3 * i + 1);
            const float c2 = __builtin_nontemporal_load(rgb + 3 * i + 2);
            acc_r0 += vw * c0; acc_r1 += vw * c1; acc_r2 += vw * c2;

            float n0 = __builtin_nontemporal_load(nablas + 3 * i + 0);
            float n1 = __builtin_nontemporal_load(nablas + 3 * i + 1);
            float n2 = __builtin_nontemporal_load(nablas + 3 * i + 2);
            n0 = fminf(fmaxf(n0, -1.f), 1.f);
            n1 = fminf(fmaxf(n1, -1.f), 1.f);
            n2 = fminf(fmaxf(n2, -1.f), 1.f);
            const float nrm = sqrtf(n0 * n0 + n1 * n1 + n2 * n2);
            const float inv = vw / fmaxf(nrm, EPS_NORM);
            acc_n0 += n0 * inv; acc_n1 += n1 * inv; acc_n2 += n2 * inv;
        }
    }

    // Butterfly reductions across the wave (all lanes end with the totals).
    #pragma unroll
    for (int off = 16; off > 0; off >>= 1) {
        acc_vw  += __shfl_xor(acc_vw,  off, 32);
        acc_vwt += __shfl_xor(acc_vwt, off, 32);
        acc_r0  += __shfl_xor(acc_r0,  off, 32);
        acc_r1  += __shfl_xor(acc_r1,  off, 32);
        acc_r2  += __shfl_xor(acc_r2,  off, 32);
        acc_n0  += __shfl_xor(acc_n0,  off, 32);
        acc_n1  += __shfl_xor(acc_n1,  off, 32);
        acc_n2  += __shfl_xor(acc_n2,  off, 32);
    }

    if (lane == 0) {
        const int ri = ray_inds_hit[r];
        // depth = sum(vw/(vw_sum+eps) * t) == sum(vw*t)/(vw_sum+eps)
        const float depth = acc_vwt / (acc_vw + EPS_DIV);
        float4* o = reinterpret_cast<float4*>(out + (size_t)ri * 8);
        o[0] = make_float4(acc_vw, depth, acc_r0, acc_r1);
        o[1] = make_float4(acc_r2, acc_n0, acc_n1, acc_n2);
    }
}

// ---------------------------------------------------------------------------
// Host launcher. Inputs (setup_inputs order):
//   0 opacity_alpha [P] f32      1 t [P] f32
//   2 rgb [P,3] f32              3 nablas_in_world [P,3] f32
//   4 ray_inds_hit [N_HIT] int   5 segment_ids [P] int (unused; starts suffice)
//   6 starts [N_HIT] int         7 num_rays [1] int (unused; out_size/8)
// ---------------------------------------------------------------------------
extern "C" void kernel_launch(void* const* d_in, const int* in_sizes, int n_in,
                              void* d_out, int out_size, void* d_ws, size_t ws_size,
                              hipStream_t stream) {
    const float* alpha   = (const float*)d_in[0];
    const float* tdep    = (const float*)d_in[1];
    const float* rgb     = (const float*)d_in[2];
    const float* nablas  = (const float*)d_in[3];
    const int*   rayinds = (const int*)d_in[4];
    const int*   starts  = (const int*)d_in[6];

    const int P     = in_sizes[0];
    const int n_hit = in_sizes[4];

    float* out = (float*)d_out;

    // 1) zero the full [N_rays, 8] output (float4 stores; out_size % 4 == 0)
    const int n4 = out_size / 4;
    vr_zero_kernel<<<(n4 + 255) / 256, 256, 0, stream>>>((float4*)out, n4);

    // 2) one wave per hit ray; 8 waves (256 threads) per block
    const int blocks = (n_hit + 7) / 8;
    vr_render_kernel<<<blocks, 256, 0, stream>>>(alpha, tdep, rgb, nablas,
                                                 rayinds, starts, out, n_hit, P);
}